// MultiHeadAttention_79130477461654
// MI455X (gfx1250) — compile-verified
//
#include <hip/hip_runtime.h>
#include <stddef.h>

// ---------------------------------------------------------------------------
// MultiHeadAttention (N=8, S=1024, E=1024, H=16, D=64) — fp32
//
// The reference einsum 'nhqk,nhvd->nhqd' contracts k and v INDEPENDENTLY:
//   out[n,h,q,d] = (sum_k softmax(..)[n,h,q,k]) * (sum_s V[n,h,s,d])
//                = 1 * sum_s V[n,h,s,d]                  (q-independent!)
// and sum_s V[n] = (sum_s x[n,s,:]) @ Wv.  Hence:
//   row[n,:]   = ((sum_s x[n,s,:]) @ Wv) @ Wo + bo
//   out[n,s,:] = row[n,:]   for every s
// Traffic: 32 MB read (x) + 32 MB write (out) => ~2.75 us @ 23.3 TB/s.
// The two 16x1024x1024 GEMMs run on the native fp32 tensor path
// V_WMMA_F32_16X16X4_F32 (full fp32 precision, M padded 8->16).
// ---------------------------------------------------------------------------

typedef __attribute__((ext_vector_type(2))) float v2f;
typedef __attribute__((ext_vector_type(4))) float f4;
typedef __attribute__((ext_vector_type(8))) float v8f;

#define NB       8      // batch
#define SEQ      1024   // sequence length
#define EMB      1024   // embed dim
#define SCHUNKS  16     // s-chunks for deterministic two-stage reduction
#define SCLEN    (SEQ / SCHUNKS)   // 64

// ---- Stage 1a: partial column sums of x over s ----------------------------
// grid (NB, SCHUNKS), block 256.  thread t owns float4 at e = 4t.
__global__ void colsum_partial(const float* __restrict__ x,
                               float* __restrict__ partial) {
  const int n  = blockIdx.x;
  const int sc = blockIdx.y;
  const int t  = threadIdx.x;                 // 0..255
  const f4* __restrict__ xv =
      reinterpret_cast<const f4*>(x + (size_t)n * SEQ * EMB) + t;
  f4 acc = {0.f, 0.f, 0.f, 0.f};
  const int s0 = sc * SCLEN;
#pragma unroll 4
  for (int s = 0; s < SCLEN; ++s) {
    acc += xv[(size_t)(s0 + s) * (EMB / 4)];
  }
  reinterpret_cast<f4*>(partial)[((size_t)n * SCHUNKS + sc) * (EMB / 4) + t] = acc;
}

// ---- Stage 1b: reduce partials -> Xsum[16][EMB] (rows 8..15 zero pad) -----
// grid (16), block 256.
__global__ void colsum_final(const float* __restrict__ partial,
                             float* __restrict__ xsum) {
  const int m = blockIdx.x;                   // 0..15
  const int t = threadIdx.x;
  f4 acc = {0.f, 0.f, 0.f, 0.f};
  if (m < NB) {
    const f4* __restrict__ pv =
        reinterpret_cast<const f4*>(partial) + (size_t)m * SCHUNKS * (EMB / 4) + t;
#pragma unroll
    for (int c = 0; c < SCHUNKS; ++c) acc += pv[(size_t)c * (EMB / 4)];
  }
  reinterpret_cast<f4*>(xsum)[(size_t)m * (EMB / 4) + t] = acc;
}

// ---- Stage 2: C[16][1024] = A[16][K] @ B[K][1024] (+bias), fp32 WMMA ------
// One wave per 16x16 output tile; 64 tiles; grid 16 x blockDim 128 (4 waves).
// A-matrix lane layout (16x4 f32): lane l -> row m=l%16, k-pair kb=(l/16)*2.
// B-matrix lane layout (4x16 f32): lane l -> col n=l%16, k-pair kb=(l/16)*2.
// C/D: VGPR j: lanes 0-15 row j, lanes 16-31 row j+8.
__global__ void gemm16_wmma(const float* __restrict__ A,    // 16 x K row-major
                            const float* __restrict__ B,    // K x 1024 row-major
                            const float* __restrict__ bias, // [1024] or nullptr
                            float* __restrict__ C,          // 16 x 1024 row-major
                            int K) {
  const int lane = threadIdx.x & 31;
  const int wave = threadIdx.x >> 5;
  const int tile = blockIdx.x * (blockDim.x >> 5) + wave;   // 0..63
  const int n0   = tile * 16;
  const int m    = lane & 15;                 // A row / B column within tile
  const int kb   = (lane >> 4) << 1;          // 0 or 2

  v8f c = {};
  for (int k = 0; k < K; k += 4) {
    // A: two consecutive fp32 (8B-aligned) from row m
    v2f a = *reinterpret_cast<const v2f*>(A + (size_t)m * K + k + kb);
    // B: column n0+m, rows k+kb and k+kb+1
    v2f b;
    b.x = B[(size_t)(k + kb)     * EMB + n0 + m];
    b.y = B[(size_t)(k + kb + 1) * EMB + n0 + m];
    // D = A*B + C   (emits v_wmma_f32_16x16x4_f32)
    c = __builtin_amdgcn_wmma_f32_16x16x4_f32(
        /*neg_a=*/false, a, /*neg_b=*/false, b,
        /*c_mod=*/(short)0, c, /*reuse_a=*/false, /*reuse_b=*/false);
  }

  const int rbase = (lane < 16) ? 0 : 8;      // lanes 16-31 hold rows 8..15
  const int col   = n0 + m;
  const float bv  = bias ? bias[col] : 0.f;
#pragma unroll
  for (int j = 0; j < 8; ++j) {
    C[(size_t)(rbase + j) * EMB + col] = c[j] + bv;
  }
}

// ---- Stage 3: broadcast row[n] across all s positions ---------------------
// grid (SEQ, NB), block 256, float4 stores (32 MB streaming write).
__global__ void broadcast_out(const float* __restrict__ rowbuf,  // 16 x EMB (rows 0..7 valid)
                              float* __restrict__ out) {
  const int s = blockIdx.x;
  const int n = blockIdx.y;
  const int t = threadIdx.x;
  const f4 v = reinterpret_cast<const f4*>(rowbuf)[(size_t)n * (EMB / 4) + t];
  reinterpret_cast<f4*>(out)[((size_t)n * SEQ + s) * (EMB / 4) + t] = v;
}

// ---------------------------------------------------------------------------
extern "C" void kernel_launch(void* const* d_in, const int* in_sizes, int n_in,
                              void* d_out, int out_size, void* d_ws, size_t ws_size,
                              hipStream_t stream) {
  (void)in_sizes; (void)n_in; (void)out_size; (void)ws_size;

  const float* x  = (const float*)d_in[0];
  // d_in[1] = Wq, d_in[2] = Wk : mathematically irrelevant (softmax rows sum to 1)
  const float* Wv = (const float*)d_in[3];
  const float* Wo = (const float*)d_in[4];
  const float* bo = (const float*)d_in[5];
  float* out = (float*)d_out;

  float* ws      = (float*)d_ws;
  float* partial = ws;                               // NB*SCHUNKS*EMB = 128K floats
  float* xsum    = partial + NB * SCHUNKS * EMB;     // 16*EMB
  float* vsum    = xsum + 16 * EMB;                  // 16*EMB
  float* rowbuf  = vsum + 16 * EMB;                  // 16*EMB   (total ~704 KB)

  // 1) Xsum[n,:] = sum_s x[n,s,:]   (deterministic two-stage reduction)
  colsum_partial<<<dim3(NB, SCHUNKS), 256, 0, stream>>>(x, partial);
  colsum_final<<<16, 256, 0, stream>>>(partial, xsum);

  // 2) Vsum = Xsum @ Wv ; row = Vsum @ Wo + bo     (fp32 WMMA)
  gemm16_wmma<<<16, 128, 0, stream>>>(xsum, Wv, nullptr, vsum, EMB);
  gemm16_wmma<<<16, 128, 0, stream>>>(vsum, Wo, bo, rowbuf, EMB);

  // 3) out[n,s,:] = row[n,:]
  broadcast_out<<<dim3(SEQ, NB), 256, 0, stream>>>(rowbuf, out);
}